// GAT_41154376631082
// MI455X (gfx1250) — compile-verified
//
#include <hip/hip_runtime.h>
#include <hip/hip_bf16.h>
#include <math.h>

#define N_NODES   50000
#define F_IN      256
#define NHEAD1    8
#define CH1       32
#define F_MID     256          // NHEAD1*CH1
#define CH2       32
#define E_EDGES   500000
#define E_TOT     (E_EDGES + N_NODES)   // with self-loops
#define NEG_SLOPE 0.2f

typedef __attribute__((ext_vector_type(2))) float v2f;
typedef __attribute__((ext_vector_type(8))) float v8f;

// ---------------------------------------------------------------------------
// f32 WMMA GEMM:  C[MxN] = A[MxK] * B[KxN], row-major, M%16==0, N%16==0, K%4==0
// One wave32 per 16x16 output tile, K swept 4 at a time via V_WMMA_F32_16X16X4_F32
// Fragment layouts per CDNA5 ISA 7.12.2:
//   A 16x4 : lanes 0-15 hold M=lane K={0,1} in v0/v1; lanes 16-31 hold K={2,3}
//   B 4x16 : mirrored (N takes the role of M)
//   C 16x16: vgpr r holds row r (lanes 0-15) and row r+8 (lanes 16-31), col=lane&15
// ---------------------------------------------------------------------------
__global__ void gemm_wmma_f32(const float* __restrict__ A,
                              const float* __restrict__ B,
                              float* __restrict__ C,
                              int M, int N, int K) {
    const int lane = threadIdx.x & 31;
    const int wave = threadIdx.x >> 5;
    const int wavesPerBlock = blockDim.x >> 5;
    const int tile = blockIdx.x * wavesPerBlock + wave;
    const int ntn = N >> 4;
    const int tm = tile / ntn;
    const int tn = tile % ntn;
    if (tm * 16 >= M) return;               // wave-uniform guard (EXEC stays all-1 for WMMA)
    const int m0 = tm * 16, n0 = tn * 16;
    const int lo = lane & 15;               // M (A) / N (B,C) index within tile
    const int hi = lane >> 4;               // selects K pair (A,B) / row half (C)

    v8f acc = {};
    for (int k0 = 0; k0 < K; k0 += 4) {
        v2f a, b;
        const float* ap = A + (size_t)(m0 + lo) * K + k0 + hi * 2;
        a.x = ap[0];
        a.y = ap[1];
        const float* bp = B + (size_t)(k0 + hi * 2) * N + (n0 + lo);
        b.x = bp[0];
        b.y = bp[N];
        acc = __builtin_amdgcn_wmma_f32_16x16x4_f32(false, a, false, b,
                                                    (short)0, acc, false, false);
    }
    float* cp = C + (size_t)(m0 + hi * 8) * N + n0 + lo;
    #pragma unroll
    for (int r = 0; r < 8; ++r)
        cp[(size_t)r * N] = acc[r];
}

// ---------------------------------------------------------------------------
__global__ void fill_f32(float* __restrict__ p, float v, int n) {
    int i = blockIdx.x * blockDim.x + threadIdx.x;
    if (i < n) p[i] = v;
}

// per-(node,head) attention logits: as[n,h] = <h[n,h,:], a_src[h,:]>, ad likewise
__global__ void node_alpha(const float* __restrict__ h,
                           const float* __restrict__ a_src,
                           const float* __restrict__ a_dst,
                           float* __restrict__ as, float* __restrict__ ad,
                           int n, int Hh, int Cc) {
    int i = blockIdx.x * blockDim.x + threadIdx.x;
    if (i >= n * Hh) return;
    int node = i / Hh, hd = i % Hh;
    const float* hp = h + (size_t)node * (Hh * Cc) + hd * Cc;
    const float* sp = a_src + hd * Cc;
    const float* dp = a_dst + hd * Cc;
    float s = 0.f, d = 0.f;
    for (int c = 0; c < Cc; ++c) { s += hp[c] * sp[c]; d += hp[c] * dp[c]; }
    as[i] = s; ad[i] = d;
}

__device__ __forceinline__ void edge_sd(const long long* __restrict__ ei,
                                        int e, int& s, int& d) {
    if (e < E_EDGES) { s = (int)ei[e]; d = (int)ei[E_EDGES + e]; }
    else             { s = d = e - E_EDGES; }          // appended self-loops
}

// two-sided float atomic max (valid under mixed signs)
__device__ __forceinline__ void atomicMaxF(float* addr, float val) {
    if (val >= 0.f) atomicMax((int*)addr, __float_as_int(val));
    else            atomicMin((unsigned int*)addr, __float_as_uint(val));
}

__device__ __forceinline__ float leaky(float v) {
    return v > 0.f ? v : NEG_SLOPE * v;
}

__global__ void edge_max(const long long* __restrict__ ei,
                         const float* __restrict__ as, const float* __restrict__ ad,
                         float* __restrict__ m, int Hh) {
    int i = blockIdx.x * blockDim.x + threadIdx.x;
    if (i >= E_TOT * Hh) return;
    int e = i / Hh, hd = i % Hh;
    int s, d; edge_sd(ei, e, s, d);
    float ev = leaky(as[s * Hh + hd] + ad[d * Hh + hd]);
    atomicMaxF(&m[d * Hh + hd], ev);
}

__global__ void edge_expsum(const long long* __restrict__ ei,
                            const float* __restrict__ as, const float* __restrict__ ad,
                            const float* __restrict__ m, float* __restrict__ den, int Hh) {
    int i = blockIdx.x * blockDim.x + threadIdx.x;
    if (i >= E_TOT * Hh) return;
    int e = i / Hh, hd = i % Hh;
    int s, d; edge_sd(ei, e, s, d);
    float ev = leaky(as[s * Hh + hd] + ad[d * Hh + hd]);
    float mx = m[d * Hh + hd];
    if (!isfinite(mx)) mx = 0.f;
    atomicAdd(&den[d * Hh + hd], expf(ev - mx));
}

// layer-1 weighted scatter: one 256-thread block per edge (8 heads x 32 ch)
__global__ void edge_msg1(const long long* __restrict__ ei,
                          const float* __restrict__ h,
                          const float* __restrict__ as, const float* __restrict__ ad,
                          const float* __restrict__ m, const float* __restrict__ den,
                          float* __restrict__ acc) {
    __shared__ float s_w[NHEAD1];
    int e = blockIdx.x;
    int t = threadIdx.x;              // 0..255
    int hd = t >> 5;
    int s, d; edge_sd(ei, e, s, d);
    if ((t & 31) == 0) {
        float ev = leaky(as[s * NHEAD1 + hd] + ad[d * NHEAD1 + hd]);
        float mx = m[d * NHEAD1 + hd];
        if (!isfinite(mx)) mx = 0.f;
        s_w[hd] = expf(ev - mx) / (den[d * NHEAD1 + hd] + 1e-16f);
    }
    __syncthreads();
    float w = s_w[hd];
    atomicAdd(&acc[(size_t)d * F_MID + t], w * h[(size_t)s * F_MID + t]);
}

// layer-2 weighted scatter: one thread per (edge, channel), H=1
__global__ void edge_msg2(const long long* __restrict__ ei,
                          const float* __restrict__ h,
                          const float* __restrict__ as, const float* __restrict__ ad,
                          const float* __restrict__ m, const float* __restrict__ den,
                          float* __restrict__ acc) {
    int i = blockIdx.x * blockDim.x + threadIdx.x;
    if (i >= E_TOT * CH2) return;
    int e = i / CH2, c = i % CH2;
    int s, d; edge_sd(ei, e, s, d);
    float ev = leaky(as[s] + ad[d]);
    float mx = m[d];
    if (!isfinite(mx)) mx = 0.f;
    float w = expf(ev - mx) / (den[d] + 1e-16f);
    atomicAdd(&acc[(size_t)d * CH2 + c], w * h[(size_t)s * CH2 + c]);
}

__global__ void elu_bias(float* __restrict__ acc, const float* __restrict__ b, int n) {
    int i = blockIdx.x * blockDim.x + threadIdx.x;
    if (i >= n * F_MID) return;
    float v = acc[i] + b[i % F_MID];
    acc[i] = v > 0.f ? v : (expf(v) - 1.f);
}

// bias + log_softmax over 32 classes: one wave32 lane per class
__global__ void finalize_logsoftmax(const float* __restrict__ acc,
                                    const float* __restrict__ b,
                                    float* __restrict__ out) {
    int i = blockIdx.x * blockDim.x + threadIdx.x;
    if (i >= N_NODES * CH2) return;
    int c = threadIdx.x & 31;
    float v = acc[i] + b[c];
    float mx = v;
    #pragma unroll
    for (int off = 16; off > 0; off >>= 1)
        mx = fmaxf(mx, __shfl_xor(mx, off, 32));
    float ex = expf(v - mx);
    float sum = ex;
    #pragma unroll
    for (int off = 16; off > 0; off >>= 1)
        sum += __shfl_xor(sum, off, 32);
    out[i] = v - mx - logf(sum);
}

// ---------------------------------------------------------------------------
extern "C" void kernel_launch(void* const* d_in, const int* in_sizes, int n_in,
                              void* d_out, int out_size, void* d_ws, size_t ws_size,
                              hipStream_t stream) {
    const float*     x    = (const float*)d_in[0];
    const long long* ei   = (const long long*)d_in[1];   // int64 edge_index [2,E]
    const float*     W1   = (const float*)d_in[2];
    const float*     a_s1 = (const float*)d_in[3];
    const float*     a_d1 = (const float*)d_in[4];
    const float*     b1   = (const float*)d_in[5];
    const float*     W2   = (const float*)d_in[6];
    const float*     a_s2 = (const float*)d_in[7];
    const float*     a_d2 = (const float*)d_in[8];
    const float*     b2   = (const float*)d_in[9];
    float* out = (float*)d_out;

    float* ws  = (float*)d_ws;
    float* h1   = ws; ws += (size_t)N_NODES * F_MID;   // layer-1 features
    float* acc1 = ws; ws += (size_t)N_NODES * F_MID;   // layer-1 aggregate / elu in-place
    float* as1  = ws; ws += (size_t)N_NODES * NHEAD1;
    float* ad1  = ws; ws += (size_t)N_NODES * NHEAD1;
    float* m1   = ws; ws += (size_t)N_NODES * NHEAD1;
    float* d1   = ws; ws += (size_t)N_NODES * NHEAD1;
    float* h2   = ws; ws += (size_t)N_NODES * CH2;
    float* acc2 = ws; ws += (size_t)N_NODES * CH2;
    float* as2  = ws; ws += (size_t)N_NODES;
    float* ad2  = ws; ws += (size_t)N_NODES;
    float* m2   = ws; ws += (size_t)N_NODES;
    float* d2   = ws; ws += (size_t)N_NODES;

    const int TPB = 256;
    const int WPB = 4;                       // waves per block in GEMM

    // ---------------- Layer 1 ----------------
    {
        int tiles = (N_NODES / 16) * (F_MID / 16);       // 50000 tiles
        gemm_wmma_f32<<<(tiles + WPB - 1) / WPB, 32 * WPB, 0, stream>>>(
            x, W1, h1, N_NODES, F_MID, F_IN);
    }
    node_alpha<<<(N_NODES * NHEAD1 + TPB - 1) / TPB, TPB, 0, stream>>>(
        h1, a_s1, a_d1, as1, ad1, N_NODES, NHEAD1, CH1);
    fill_f32<<<((int)((size_t)N_NODES * F_MID) + TPB - 1) / TPB, TPB, 0, stream>>>(
        acc1, 0.f, N_NODES * F_MID);
    fill_f32<<<(N_NODES * NHEAD1 + TPB - 1) / TPB, TPB, 0, stream>>>(
        m1, -INFINITY, N_NODES * NHEAD1);
    fill_f32<<<(N_NODES * NHEAD1 + TPB - 1) / TPB, TPB, 0, stream>>>(
        d1, 0.f, N_NODES * NHEAD1);
    edge_max<<<(E_TOT * NHEAD1 + TPB - 1) / TPB, TPB, 0, stream>>>(ei, as1, ad1, m1, NHEAD1);
    edge_expsum<<<(E_TOT * NHEAD1 + TPB - 1) / TPB, TPB, 0, stream>>>(ei, as1, ad1, m1, d1, NHEAD1);
    edge_msg1<<<E_TOT, 256, 0, stream>>>(ei, h1, as1, ad1, m1, d1, acc1);
    elu_bias<<<((int)((size_t)N_NODES * F_MID) + TPB - 1) / TPB, TPB, 0, stream>>>(
        acc1, b1, N_NODES);

    // ---------------- Layer 2 ----------------
    {
        int tiles = (N_NODES / 16) * (CH2 / 16);         // 6250 tiles
        gemm_wmma_f32<<<(tiles + WPB - 1) / WPB, 32 * WPB, 0, stream>>>(
            acc1, W2, h2, N_NODES, CH2, F_MID);
    }
    node_alpha<<<(N_NODES + TPB - 1) / TPB, TPB, 0, stream>>>(
        h2, a_s2, a_d2, as2, ad2, N_NODES, 1, CH2);
    fill_f32<<<(N_NODES * CH2 + TPB - 1) / TPB, TPB, 0, stream>>>(acc2, 0.f, N_NODES * CH2);
    fill_f32<<<(N_NODES + TPB - 1) / TPB, TPB, 0, stream>>>(m2, -INFINITY, N_NODES);
    fill_f32<<<(N_NODES + TPB - 1) / TPB, TPB, 0, stream>>>(d2, 0.f, N_NODES);
    edge_max<<<(E_TOT + TPB - 1) / TPB, TPB, 0, stream>>>(ei, as2, ad2, m2, 1);
    edge_expsum<<<(E_TOT + TPB - 1) / TPB, TPB, 0, stream>>>(ei, as2, ad2, m2, d2, 1);
    edge_msg2<<<(E_TOT * CH2 + TPB - 1) / TPB, TPB, 0, stream>>>(ei, h2, as2, ad2, m2, d2, acc2);
    finalize_logsoftmax<<<(N_NODES * CH2 + TPB - 1) / TPB, TPB, 0, stream>>>(acc2, b2, out);

    (void)in_sizes; (void)n_in; (void)out_size; (void)ws_size;
}